// RegionEmbedding_49589692400149
// MI455X (gfx1250) — compile-verified
//
#include <hip/hip_runtime.h>

// WMMA fragment types (wave32):
//  A 16x4 f32  -> 2 VGPRs/lane ; B 4x16 f32 -> 2 VGPRs/lane ; C/D 16x16 -> 8 VGPRs/lane
typedef float v2f __attribute__((ext_vector_type(2)));
typedef float v4f __attribute__((ext_vector_type(4)));
typedef float v8f __attribute__((ext_vector_type(8)));
typedef unsigned int v4u __attribute__((ext_vector_type(4)));
typedef unsigned int v8u __attribute__((ext_vector_type(8)));

#define LDS_STRIDE 68              // 64 + 4 dword pad -> conflict-free strided ds_load_b64
#define TILE_FLOATS (64 * LDS_STRIDE)

__global__ void init_out_kernel(float* out, int n) {
    int i = blockIdx.x * blockDim.x + threadIdx.x;
    if (i < n) out[i] = -__builtin_inff();
}

__device__ inline void atomic_max_f32(float* addr, float val) {
    int* ia = reinterpret_cast<int*>(addr);
    int old = *ia;
    while (__int_as_float(old) < val) {
        int assumed = old;
        old = atomicCAS(ia, assumed, __float_as_int(val));
        if (old == assumed) break;
    }
}

// TDM: DMA one contiguous 64x64 f32 tile (16 KB) global -> LDS.
// D# group1 pad feature inserts 4 dwords after every 64 dwords, giving the
// 68-dword padded row stride the WMMA fragment reads expect.
__device__ __forceinline__ void tdm_load_tile(unsigned lds_byte_addr, const float* gsrc) {
    unsigned long long ga = (unsigned long long)(uintptr_t)gsrc;
    v4u g0;
    g0.x = 1u;                                                  // count=1 (valid user D#)
    g0.y = lds_byte_addr;                                       // lds_addr
    g0.z = (unsigned)ga;                                        // global_addr[31:0]
    g0.w = ((unsigned)(ga >> 32) & 0x01FFFFFFu) | 0x80000000u;  // global_addr[56:32] | type=2
    v8u g1;
    g1[0] = (2u << 16)                 // data_size = 4 bytes
          | (1u << 20)                 // pad_enable
          | (5u << 22)                 // pad_interval: every 64 dwords
          | (3u << 25);                // pad_amount: 4 dwords
    g1[1] = (4096u & 0xFFFFu) << 16;   // tensor_dim0[15:0] = 4096
    g1[2] = (4096u >> 16) | (1u << 16);// tensor_dim0[31:16]=0, tensor_dim1=1
    g1[3] = (4096u << 16);             // tile_dim0 = 4096 (1-D contiguous tile)
    g1[4] = 0u;                        // tile_dim1 = 0 (unused), tile_dim2 = 0
    g1[5] = 4096u;                     // tensor_dim0_stride[31:0]
    g1[6] = 0u;
    g1[7] = 0u;
    asm volatile("tensor_load_to_lds %0, %1" :: "s"(g0), "s"(g1) : "memory");
}

__device__ __forceinline__ void wait_tensorcnt0() {
#if __has_builtin(__builtin_amdgcn_s_wait_tensorcnt)
    __builtin_amdgcn_s_wait_tensorcnt(0);
#else
    asm volatile("s_wait_tensorcnt 0x0" ::: "memory");
#endif
}
__device__ __forceinline__ void wait_tensorcnt2() {
#if __has_builtin(__builtin_amdgcn_s_wait_tensorcnt)
    __builtin_amdgcn_s_wait_tensorcnt(2);
#else
    asm volatile("s_wait_tensorcnt 0x2" ::: "memory");
#endif
}

__global__ __launch_bounds__(256) void rating_max_kernel(
    const float* __restrict__ neighbor_emb,  // [B,S,64]
    const float* __restrict__ item_emb,      // [B,64]
    const float* __restrict__ K_user_item,   // [B,S,64,64]
    const float* __restrict__ K_item_user,   // [B,S,64,64]
    const float* __restrict__ weight,        // [B,S]
    float* __restrict__ out,                 // [B,64]
    int B, int S, int s_chunk)
{
    __shared__ float lds_k[2][2][TILE_FLOATS];  // [buf][0]=K_item_user, [buf][1]=K_user_item
    __shared__ float lds_nvec[2][64];           // neighbor_emb[b,s,:], double buffered
    __shared__ float lds_ivec[64];              // item_emb[b,:]
    __shared__ float lds_prof[2][64];           // per-s profiles: [0]=neighbor, [1]=item

    const int b    = blockIdx.x;
    const int s0   = blockIdx.y * s_chunk;
    const int s1   = (s0 + s_chunk < S) ? (s0 + s_chunk) : S;
    const int tid  = threadIdx.x;
    const int wave = tid >> 5;
    const int lane = tid & 31;
    const int half = lane >> 4;   // 0: lanes 0-15 hold K=0,1 ; 1: lanes 16-31 hold K=2,3
    const int mrow = lane & 15;   // M row within 16-row tile
    const int mtx  = wave >> 2;   // 0 -> neighbor chain (K_item_user), 1 -> item chain (K_user_item)
    const int r_tile = (wave & 3) * 16;
    const int nIter = s1 - s0;
    if (nIter <= 0) return;

    if (tid < 16) {
        *(v4f*)&lds_ivec[tid * 4] = *(const v4f*)(item_emb + (size_t)b * 64 + tid * 4);
    }

    // Prologue: stage iteration 0 into buffer 0.
    if (wave == 0) {
        const size_t koff0 = ((size_t)(b * S + s0)) * 64 * 64;
        tdm_load_tile((unsigned)(uintptr_t)&lds_k[0][0][0], K_item_user + koff0);
        tdm_load_tile((unsigned)(uintptr_t)&lds_k[0][1][0], K_user_item + koff0);
        if (lane < 16) {
            *(v4f*)&lds_nvec[0][lane * 4] =
                *(const v4f*)(neighbor_emb + ((size_t)(b * S + s0)) * 64 + lane * 4);
        }
    }

    float runmax = -__builtin_inff();  // meaningful in tid < 64

    for (int i = 0; i < nIter; ++i) {
        const int s   = s0 + i;
        const int cur = i & 1;
        const int nxt = cur ^ 1;
        const bool have_next = (i + 1) < nIter;

        if (wave == 0) {
            if (have_next) {
                // Issue next iteration's DMA before waiting: TENSORcnt completes
                // in order, so waiting for <=2 below leaves these in flight while
                // guaranteeing the current buffer's pair is done.
                const size_t koffn = ((size_t)(b * S + s + 1)) * 64 * 64;
                tdm_load_tile((unsigned)(uintptr_t)&lds_k[nxt][0][0], K_item_user + koffn);
                tdm_load_tile((unsigned)(uintptr_t)&lds_k[nxt][1][0], K_user_item + koffn);
                if (lane < 16) {
                    *(v4f*)&lds_nvec[nxt][lane * 4] =
                        *(const v4f*)(neighbor_emb + ((size_t)(b * S + s + 1)) * 64 + lane * 4);
                }
                wait_tensorcnt2();
            } else {
                wait_tensorcnt0();
            }
        }
        __syncthreads();   // buffer `cur` is ready for all waves

        // GEMV as WMMA: D(16x16) = sum over 16 steps of A(16x4) x Bbroadcast(4x16).
        const float* rowp = &lds_k[cur][mtx][(r_tile + mrow) * LDS_STRIDE + half * 2];
        const float* vecp = (mtx == 0) ? &lds_nvec[cur][0] : lds_ivec;
        v8f acc = {0.f, 0.f, 0.f, 0.f, 0.f, 0.f, 0.f, 0.f};
        #pragma unroll
        for (int k = 0; k < 16; ++k) {
            v2f a  = *(const v2f*)(rowp + 4 * k);
            v2f bb = *(const v2f*)(vecp + 4 * k + half * 2);
            acc = __builtin_amdgcn_wmma_f32_16x16x4_f32(
                /*neg_a=*/false, a, /*neg_b=*/false, bb,
                /*c_mod=*/(short)0, acc, /*reuse_a=*/false, /*reuse_b=*/false);
        }

        // Columns of D are identical; lanes with N==0 extract their 8 rows.
        if ((lane & 15) == 0) {
            #pragma unroll
            for (int j = 0; j < 8; ++j)
                lds_prof[mtx][r_tile + half * 8 + j] = acc[j];
        }
        __syncthreads();   // profiles ready; also fences reads of buffer `cur`

        if (tid < 64) {
            float w = weight[(size_t)b * S + s];
            float rating = lds_prof[0][tid] * lds_prof[1][tid] * w;
            runmax = fmaxf(runmax, rating);
        }
    }

    if (tid < 64) {
        atomic_max_f32(out + (size_t)b * 64 + tid, runmax);
    }
}

extern "C" void kernel_launch(void* const* d_in, const int* in_sizes, int n_in,
                              void* d_out, int out_size, void* d_ws, size_t ws_size,
                              hipStream_t stream) {
    (void)n_in; (void)d_ws; (void)ws_size;
    const float* neighbor_emb = (const float*)d_in[0];
    const float* item_emb     = (const float*)d_in[1];
    const float* K_user_item  = (const float*)d_in[2];
    const float* K_item_user  = (const float*)d_in[3];
    const float* weight       = (const float*)d_in[4];
    float* out = (float*)d_out;

    const int B = in_sizes[1] / 64;          // item_emb is [B,64]
    const int S = (B > 0) ? in_sizes[4] / B  // weight is [B,S]
                          : 1;

    init_out_kernel<<<(out_size + 255) / 256, 256, 0, stream>>>(out, out_size);

    const int split   = (S >= 5) ? 5 : 1;    // B*split workgroups for occupancy
    const int s_chunk = (S + split - 1) / split;
    dim3 grid((unsigned)B, (unsigned)((S + s_chunk - 1) / s_chunk));
    rating_max_kernel<<<grid, 256, 0, stream>>>(neighbor_emb, item_emb, K_user_item,
                                                K_item_user, weight, out, B, S, s_chunk);
}